// MultiQuerySlot_6794638262553
// MI455X (gfx1250) — compile-verified
//
#include <hip/hip_runtime.h>

typedef float v2f __attribute__((ext_vector_type(2)));
typedef float v8f __attribute__((ext_vector_type(8)));

constexpr int H_ = 8, B_ = 64, S_ = 8, D_ = 64, N_ = 4096, MLPH_ = 128, ITERS_ = 3;
constexpr float LN_EPS_ = 1e-5f, EPS_ATTN_ = 1e-8f, SCALE_ = 0.125f; // D^-0.5

// D = A(16x4,f32) * B(4x16,f32) + C(16x16,f32)   -> v_wmma_f32_16x16x4_f32
__device__ __forceinline__ v8f wmma_f32(v2f a, v2f b, v8f c) {
  return __builtin_amdgcn_wmma_f32_16x16x4_f32(false, a, false, b, (short)0, c,
                                               false, false);
}

// ---------------------------------------------------------------------------
// Kernel A: x = LN(inputs); k = x@Wk; v = x@Wv.  One 16-row tile per block.
// 256 threads = 8 waves: 4 waves -> k col-tiles, 4 waves -> v col-tiles.
// ---------------------------------------------------------------------------
__global__ void __launch_bounds__(256) ln_kv_kernel(
    const float* __restrict__ inp, const float* __restrict__ lng,
    const float* __restrict__ lnb, const float* __restrict__ Wk,
    const float* __restrict__ Wv, float* __restrict__ kout,
    float* __restrict__ vout) {
  __shared__ float xln[16][68];
  __shared__ float mu[16], rstd[16];
  const int tid = threadIdx.x;
  const size_t row0 = (size_t)blockIdx.x * 16;
  const int r = tid >> 4, c0 = (tid & 15) * 4;
  float4 val = *(const float4*)(inp + (row0 + r) * D_ + c0);
  xln[r][c0] = val.x; xln[r][c0 + 1] = val.y;
  xln[r][c0 + 2] = val.z; xln[r][c0 + 3] = val.w;
  __syncthreads();
  if (tid < 16) {
    float m = 0.f;
    for (int j = 0; j < D_; ++j) m += xln[tid][j];
    m *= (1.f / D_);
    float var = 0.f;
    for (int j = 0; j < D_; ++j) { float dd = xln[tid][j] - m; var += dd * dd; }
    var *= (1.f / D_);
    mu[tid] = m; rstd[tid] = rsqrtf(var + LN_EPS_);
  }
  __syncthreads();
  {
    float m = mu[r], rs = rstd[r];
    for (int j = 0; j < 4; ++j) {
      int c = c0 + j;
      xln[r][c] = (xln[r][c] - m) * rs * lng[c] + lnb[c];
    }
  }
  __syncthreads();
  const int w = tid >> 5, lane = tid & 31, m16 = lane & 15, kh = lane >> 4;
  const float* W = (w < 4) ? Wk : Wv;
  float* outp = (w < 4) ? kout : vout;
  const int ct = (w & 3) * 16;
  v8f acc = {};
  for (int kk = 0; kk < 16; ++kk) {
    int kb = kk * 4 + kh * 2;
    v2f a; a.x = xln[m16][kb]; a.y = xln[m16][kb + 1];
    v2f b; b.x = W[kb * D_ + ct + m16]; b.y = W[(kb + 1) * D_ + ct + m16];
    acc = wmma_f32(a, b, acc);
  }
  for (int i = 0; i < 8; ++i)
    outp[(row0 + i + 8 * kh) * D_ + ct + m16] = acc[i];
}

// ---------------------------------------------------------------------------
// Kernel B: slots = mu + exp(log_sigma) * slots_init   (elementwise)
// ---------------------------------------------------------------------------
__global__ void slots_init_kernel(const float* __restrict__ sinit,
                                  const float* __restrict__ smu,
                                  const float* __restrict__ slog,
                                  float* __restrict__ slots) {
  int idx = blockIdx.x * blockDim.x + threadIdx.x;
  int d = idx & (D_ - 1);
  int h = idx / (B_ * S_ * D_);
  slots[idx] = smu[h * D_ + d] + expf(slog[h * D_ + d]) * sinit[idx];
}

// ---------------------------------------------------------------------------
// Kernel C: h_prev = LN(slots); q = h_prev @ Wq[h].  16 rows/block, 4 waves.
// ---------------------------------------------------------------------------
__global__ void __launch_bounds__(128) hprev_q_kernel(
    const float* __restrict__ slots, const float* __restrict__ lng,
    const float* __restrict__ lnb, const float* __restrict__ Wq,
    float* __restrict__ hprev, float* __restrict__ qout) {
  __shared__ float hl[16][68];
  __shared__ float mu[16], rstd[16];
  const int tid = threadIdx.x;
  const int h = blockIdx.x >> 5;      // 32 row-tiles per head (B*S/16)
  const int rt = blockIdx.x & 31;
  const int rowbase = h * (B_ * S_) + rt * 16;
  for (int e = tid; e < 16 * D_; e += 128) {
    int r = e >> 6, c = e & 63;
    hl[r][c] = slots[(size_t)(rowbase + r) * D_ + c];
  }
  __syncthreads();
  if (tid < 16) {
    float m = 0.f;
    for (int j = 0; j < D_; ++j) m += hl[tid][j];
    m *= (1.f / D_);
    float var = 0.f;
    for (int j = 0; j < D_; ++j) { float dd = hl[tid][j] - m; var += dd * dd; }
    var *= (1.f / D_);
    mu[tid] = m; rstd[tid] = rsqrtf(var + LN_EPS_);
  }
  __syncthreads();
  for (int e = tid; e < 16 * D_; e += 128) {
    int r = e >> 6, c = e & 63;
    float nv = (hl[r][c] - mu[r]) * rstd[r] * lng[h * D_ + c] + lnb[h * D_ + c];
    hl[r][c] = nv;
    hprev[(size_t)(rowbase + r) * D_ + c] = nv;
  }
  __syncthreads();
  const int w = tid >> 5, lane = tid & 31, m16 = lane & 15, kh = lane >> 4;
  const float* Wh = Wq + (size_t)h * D_ * D_;
  const int ct = w * 16;
  v8f acc = {};
  for (int kk = 0; kk < 16; ++kk) {
    int kb = kk * 4 + kh * 2;
    v2f a; a.x = hl[m16][kb]; a.y = hl[m16][kb + 1];
    v2f b; b.x = Wh[kb * D_ + ct + m16]; b.y = Wh[(kb + 1) * D_ + ct + m16];
    acc = wmma_f32(a, b, acc);
  }
  for (int i = 0; i < 8; ++i)
    qout[(size_t)(rowbase + i + 8 * kh) * D_ + ct + m16] = acc[i];
}

// ---------------------------------------------------------------------------
// Kernel D: fused attention. One block per (head-pair, batch). Single pass
// over N: logits (WMMA) -> softmax over S -> accumulate numerator (WMMA) and
// denominator; final updates = num/den. Two heads share the 16 WMMA columns.
// k/v tiles for the NEXT iteration are prefetched (global_prefetch_b8) while
// the current tile's WMMAs run — k/v are L2-resident (134 MB < 192 MB L2).
// ---------------------------------------------------------------------------
__global__ void __launch_bounds__(128) attn_kernel(
    const float* __restrict__ kbuf, const float* __restrict__ vbuf,
    const float* __restrict__ qbuf, float* __restrict__ updates) {
  __shared__ float qs[16][68];
  __shared__ float kt[64][68];
  __shared__ float at[64][17];
  __shared__ float den[16];
  const int tid = threadIdx.x;
  const int hp = blockIdx.x >> 6;   // head pair 0..3
  const int b  = blockIdx.x & 63;
  const float* kb = kbuf + (size_t)b * N_ * D_;
  const float* vb = vbuf + (size_t)b * N_ * D_;
  for (int e = tid; e < 16 * D_; e += 128) {
    int sp = e >> 6, d = e & 63;
    int h = hp * 2 + (sp >> 3), s = sp & 7;
    qs[sp][d] = qbuf[(size_t)((h * B_ + b) * S_ + s) * D_ + d];
  }
  if (tid < 16) den[tid] = 0.f;
  __syncthreads();
  const int w = tid >> 5, lane = tid & 31, m16 = lane & 15, kh = lane >> 4;
  v8f accu = {};   // wave w accumulates updates cols [16w,16w+16)
  for (int nb = 0; nb < N_; nb += 64) {
    for (int e = tid; e < 64 * D_; e += 128) {
      int r = e >> 6, c = e & 63;
      kt[r][c] = kb[(size_t)(nb + r) * D_ + c];
    }
    // prefetch next iteration's k and v tiles (16 KB each; 128 thr x 128 B)
    if (nb + 64 < N_) {
      const char* nk = (const char*)(kb + (size_t)(nb + 64) * D_);
      const char* nv = (const char*)(vb + (size_t)(nb + 64) * D_);
      __builtin_prefetch(nk + tid * 128, 0, 3);
      __builtin_prefetch(nv + tid * 128, 0, 3);
    }
    __syncthreads();
    // logits tile: rows [16w,16w+16) x 16 slot-cols  (B = q^T)
    v8f lg = {};
    for (int kk = 0; kk < 16; ++kk) {
      int kbs = kk * 4 + kh * 2;
      v2f a;  a.x = kt[w * 16 + m16][kbs]; a.y = kt[w * 16 + m16][kbs + 1];
      v2f bb; bb.x = qs[m16][kbs];         bb.y = qs[m16][kbs + 1];
      lg = wmma_f32(a, bb, lg);
    }
    for (int i = 0; i < 8; ++i)
      at[w * 16 + i + 8 * kh][m16] = lg[i] * SCALE_;
    __syncthreads();
    // softmax over S=8 per (row, head); +eps
    {
      int r = tid >> 1, hh = tid & 1;
      float mx = -3.4e38f;
      for (int s = 0; s < 8; ++s) mx = fmaxf(mx, at[r][hh * 8 + s]);
      float ex[8], sum = 0.f;
      for (int s = 0; s < 8; ++s) { ex[s] = expf(at[r][hh * 8 + s] - mx); sum += ex[s]; }
      float inv = 1.f / sum;
      for (int s = 0; s < 8; ++s) at[r][hh * 8 + s] = ex[s] * inv + EPS_ATTN_;
    }
    __syncthreads();
    if (tid < 16) {
      float dsum = 0.f;
      for (int r = 0; r < 64; ++r) dsum += at[r][tid];
      den[tid] += dsum;
    }
    // numerator: accu += attn^T (16 x 64) @ v_tile (64 x d-slice)
    for (int kk = 0; kk < 16; ++kk) {
      int kbs = kk * 4 + kh * 2;
      v2f a;  a.x = at[kbs][m16];  a.y = at[kbs + 1][m16];
      v2f bb; bb.x = vb[(size_t)(nb + kbs) * D_ + w * 16 + m16];
              bb.y = vb[(size_t)(nb + kbs + 1) * D_ + w * 16 + m16];
      accu = wmma_f32(a, bb, accu);
    }
    __syncthreads();
  }
  for (int i = 0; i < 8; ++i) {
    int sp = i + 8 * kh;
    int h = hp * 2 + (sp >> 3), s = sp & 7;
    updates[(size_t)((h * B_ + b) * S_ + s) * D_ + w * 16 + m16] = accu[i] / den[sp];
  }
}

// ---------------------------------------------------------------------------
// Kernel E: fused GRU + LayerNorm + residual MLP for a 16-row tile (one head).
// ---------------------------------------------------------------------------
__global__ void __launch_bounds__(256) gru_mlp_kernel(
    const float* __restrict__ upd, const float* __restrict__ hprev,
    const float* __restrict__ wih, const float* __restrict__ whh,
    const float* __restrict__ bih, const float* __restrict__ bhh,
    const float* __restrict__ lnmg, const float* __restrict__ lnmb,
    const float* __restrict__ w1, const float* __restrict__ b1,
    const float* __restrict__ w2, const float* __restrict__ b2,
    float* __restrict__ slots) {
  __shared__ float ut[16][68];
  __shared__ float ht[16][68];
  __shared__ float gx[16][196];
  __shared__ float gh[16][196];
  __shared__ float sn[16][68];
  __shared__ float nm[16][68];
  __shared__ float hid[16][132];
  __shared__ float mu[16], rstd[16];
  const int tid = threadIdx.x;
  const int h = blockIdx.x >> 5, rt = blockIdx.x & 31;
  const int rowbase = h * (B_ * S_) + rt * 16;
  const int r0 = tid >> 4, c00 = (tid & 15) * 4;
  {
    float4 a = *(const float4*)(upd + (size_t)(rowbase + r0) * D_ + c00);
    ut[r0][c00] = a.x; ut[r0][c00 + 1] = a.y; ut[r0][c00 + 2] = a.z; ut[r0][c00 + 3] = a.w;
    float4 c = *(const float4*)(hprev + (size_t)(rowbase + r0) * D_ + c00);
    ht[r0][c00] = c.x; ht[r0][c00 + 1] = c.y; ht[r0][c00 + 2] = c.z; ht[r0][c00 + 3] = c.w;
  }
  __syncthreads();
  const int w = tid >> 5, lane = tid & 31, m16 = lane & 15, kh = lane >> 4;
  const float* wihh = wih + (size_t)h * D_ * 192;
  const float* whhh = whh + (size_t)h * D_ * 192;
  // gx = updates@wih (12 tiles), gh = h_prev@whh (12 tiles); 8 waves x 3 jobs
  for (int T = w; T < 24; T += 8) {
    const bool isX = T < 12;
    const int ct = (T % 12) * 16;
    const float* Wt = isX ? wihh : whhh;
    v8f acc = {};
    for (int kk = 0; kk < 16; ++kk) {
      int kb = kk * 4 + kh * 2;
      v2f a;
      if (isX) { a.x = ut[m16][kb]; a.y = ut[m16][kb + 1]; }
      else     { a.x = ht[m16][kb]; a.y = ht[m16][kb + 1]; }
      v2f bb; bb.x = Wt[kb * 192 + ct + m16]; bb.y = Wt[(kb + 1) * 192 + ct + m16];
      acc = wmma_f32(a, bb, acc);
    }
    float* dst = isX ? &gx[0][0] : &gh[0][0];
    for (int i = 0; i < 8; ++i)
      dst[(i + 8 * kh) * 196 + ct + m16] = acc[i];
  }
  __syncthreads();
  {  // GRU gates
    const float* bi = bih + h * 192;
    const float* bh = bhh + h * 192;
    for (int j = 0; j < 4; ++j) {
      int d = c00 + j;
      float xr = gx[r0][d] + bi[d],        hr = gh[r0][d] + bh[d];
      float xz = gx[r0][64 + d] + bi[64 + d],  hz = gh[r0][64 + d] + bh[64 + d];
      float xn = gx[r0][128 + d] + bi[128 + d], hn = gh[r0][128 + d] + bh[128 + d];
      float rg = 1.f / (1.f + expf(-(xr + hr)));
      float zg = 1.f / (1.f + expf(-(xz + hz)));
      float ng = tanhf(xn + rg * hn);
      sn[r0][d] = (1.f - zg) * ng + zg * ht[r0][d];
    }
  }
  __syncthreads();
  if (tid < 16) {
    float m = 0.f;
    for (int j = 0; j < D_; ++j) m += sn[tid][j];
    m *= (1.f / D_);
    float var = 0.f;
    for (int j = 0; j < D_; ++j) { float dd = sn[tid][j] - m; var += dd * dd; }
    var *= (1.f / D_);
    mu[tid] = m; rstd[tid] = rsqrtf(var + LN_EPS_);
  }
  __syncthreads();
  for (int j = 0; j < 4; ++j) {
    int d = c00 + j;
    nm[r0][d] = (sn[r0][d] - mu[r0]) * rstd[r0] * lnmg[h * D_ + d] + lnmb[h * D_ + d];
  }
  __syncthreads();
  {  // MLP layer 1: relu(nm @ w1 + b1), 8 col-tiles, one per wave
    const float* w1h = w1 + (size_t)h * D_ * MLPH_;
    const float* b1h = b1 + h * MLPH_;
    const int ct = w * 16;
    v8f acc = {};
    for (int kk = 0; kk < 16; ++kk) {
      int kb = kk * 4 + kh * 2;
      v2f a;  a.x = nm[m16][kb]; a.y = nm[m16][kb + 1];
      v2f bb; bb.x = w1h[kb * MLPH_ + ct + m16]; bb.y = w1h[(kb + 1) * MLPH_ + ct + m16];
      acc = wmma_f32(a, bb, acc);
    }
    float bias = b1h[ct + m16];
    for (int i = 0; i < 8; ++i)
      hid[i + 8 * kh][ct + m16] = fmaxf(acc[i] + bias, 0.f);
  }
  __syncthreads();
  if (w < 4) {  // MLP layer 2 + residual, 4 col-tiles
    const float* w2h = w2 + (size_t)h * MLPH_ * D_;
    const float* b2h = b2 + h * D_;
    const int ct = w * 16;
    v8f acc = {};
    for (int kk = 0; kk < 32; ++kk) {
      int kb = kk * 4 + kh * 2;
      v2f a;  a.x = hid[m16][kb]; a.y = hid[m16][kb + 1];
      v2f bb; bb.x = w2h[kb * D_ + ct + m16]; bb.y = w2h[(kb + 1) * D_ + ct + m16];
      acc = wmma_f32(a, bb, acc);
    }
    float bias = b2h[ct + m16];
    for (int i = 0; i < 8; ++i) {
      int rr = i + 8 * kh;
      slots[(size_t)(rowbase + rr) * D_ + ct + m16] = sn[rr][ct + m16] + acc[i] + bias;
    }
  }
}

__global__ void copy_out_kernel(const float* __restrict__ slots,
                                float* __restrict__ out) {
  int i = blockIdx.x * blockDim.x + threadIdx.x;
  if (i < B_ * S_ * D_) out[i] = slots[i];   // head 0 block
}

extern "C" void kernel_launch(void* const* d_in, const int* in_sizes, int n_in,
                              void* d_out, int out_size, void* d_ws, size_t ws_size,
                              hipStream_t stream) {
  (void)in_sizes; (void)n_in; (void)out_size; (void)ws_size;
  const float* inputs  = (const float*)d_in[0];
  const float* sinit   = (const float*)d_in[1];
  const float* smu     = (const float*)d_in[2];
  const float* slog    = (const float*)d_in[3];
  const float* ln_in_g = (const float*)d_in[4];
  const float* ln_in_b = (const float*)d_in[5];
  const float* Wk      = (const float*)d_in[6];
  const float* Wv      = (const float*)d_in[7];
  const float* ln_s_g  = (const float*)d_in[8];
  const float* ln_s_b  = (const float*)d_in[9];
  const float* Wq      = (const float*)d_in[10];
  const float* gru_wih = (const float*)d_in[11];
  const float* gru_whh = (const float*)d_in[12];
  const float* gru_bih = (const float*)d_in[13];
  const float* gru_bhh = (const float*)d_in[14];
  const float* ln_m_g  = (const float*)d_in[15];
  const float* ln_m_b  = (const float*)d_in[16];
  const float* mlp_w1  = (const float*)d_in[17];
  const float* mlp_b1  = (const float*)d_in[18];
  const float* mlp_w2  = (const float*)d_in[19];
  const float* mlp_b2  = (const float*)d_in[20];

  float* ws = (float*)d_ws;
  float* kbuf  = ws;
  float* vbuf  = kbuf + (size_t)B_ * N_ * D_;
  float* slots = vbuf + (size_t)B_ * N_ * D_;
  float* hprev = slots + H_ * B_ * S_ * D_;
  float* qbuf  = hprev + H_ * B_ * S_ * D_;
  float* updv  = qbuf + H_ * B_ * S_ * D_;

  ln_kv_kernel<<<(B_ * N_) / 16, 256, 0, stream>>>(inputs, ln_in_g, ln_in_b,
                                                   Wk, Wv, kbuf, vbuf);
  slots_init_kernel<<<(H_ * B_ * S_ * D_) / 256, 256, 0, stream>>>(sinit, smu,
                                                                   slog, slots);
  for (int it = 0; it < ITERS_; ++it) {
    hprev_q_kernel<<<H_ * 32, 128, 0, stream>>>(slots, ln_s_g, ln_s_b, Wq,
                                                hprev, qbuf);
    attn_kernel<<<(H_ / 2) * B_, 128, 0, stream>>>(kbuf, vbuf, qbuf, updv);
    gru_mlp_kernel<<<H_ * 32, 256, 0, stream>>>(updv, hprev, gru_wih, gru_whh,
                                                gru_bih, gru_bhh, ln_m_g, ln_m_b,
                                                mlp_w1, mlp_b1, mlp_w2, mlp_b2,
                                                slots);
  }
  copy_out_kernel<<<(B_ * S_ * D_ + 255) / 256, 256, 0, stream>>>(slots,
                                                                  (float*)d_out);
}